// HiddenConv_3513283248868
// MI455X (gfx1250) — compile-verified
//
#include <hip/hip_runtime.h>
#include <hip/hip_bf16.h>

// GCN / VGAE encoder for MI455X (gfx1250), wave32 + WMMA f32 path.
// N=100000, F_IN=256, H1=128, H2=64, E=1600000.

typedef float v2f __attribute__((ext_vector_type(2)));
typedef float v8f __attribute__((ext_vector_type(8)));

#define GEMM_BLOCK 320          // 10 waves; 6250 row-tiles = 625 * 10 (exact)
#define WAVES_PER_BLOCK 10
#define MAX_K 256

// ---------------------------------------------------------------------------
// 0) init: zero agg1 + d_out, set deg=1.0 (self-loop) — one pass over N*128
// ---------------------------------------------------------------------------
__global__ void k_init(float* __restrict__ agg1, float* __restrict__ out,
                       float* __restrict__ deg, int n_nodes, long long total) {
  long long i = (long long)blockIdx.x * blockDim.x + threadIdx.x;
  if (i >= total) return;
  agg1[i] = 0.0f;
  out[i]  = 0.0f;              // d_out is N*64 (mu) + N*64 (logvar) = N*128
  if (i < n_nodes) deg[i] = 1.0f;
}

// ---------------------------------------------------------------------------
// 1) degree accumulate over edges, then rsqrt in place
// ---------------------------------------------------------------------------
__global__ void k_deg(const int* __restrict__ dst, float* __restrict__ deg,
                      long long nE) {
  long long e = (long long)blockIdx.x * blockDim.x + threadIdx.x;
  if (e >= nE) return;
  atomicAdd(&deg[dst[e]], 1.0f);
}

__global__ void k_rsqrt(float* __restrict__ deg, int n_nodes) {
  int i = blockIdx.x * blockDim.x + threadIdx.x;
  if (i >= n_nodes) return;
  deg[i] = __frsqrt_rn(deg[i]);   // deg buffer now holds dinv_sqrt
}

// ---------------------------------------------------------------------------
// 2) GEMM via V_WMMA_F32_16X16X4_F32, LDS-staged B panel.
//    Block = 10 waves sharing one 16-column B panel; each wave owns one
//    16x16 C tile. B panel stored as interleaved K-pairs so each lane's
//    per-step fragment is one aligned ds_load_b64:
//       Bs[p*32 + n*2 + j] = B[2p+j][col0+n]   (p = k/2, j in {0,1})
//    A frag (16x4): lane m=l&15, half=l>>4 holds {A[m][k+2h], A[m][k+2h+1]}.
//    D: VGPR v, lane l -> C[v + 8*(l>>4)][l&15].
// ---------------------------------------------------------------------------
__global__ void k_gemm_wmma(const float* __restrict__ A,
                            const float* __restrict__ B,
                            float* __restrict__ C,
                            int M, int K, int Nout) {
  __shared__ float Bs[MAX_K * 16];           // 16 KB max (K=256)

  const int colTiles = Nout >> 4;
  const int colTile  = blockIdx.x % colTiles;
  const int rowGroup = blockIdx.x / colTiles;
  const int wave     = threadIdx.x >> 5;
  const int lane     = threadIdx.x & 31;
  const int col0     = colTile << 4;

  // cooperative B-panel fill: K*16 floats, interleaved K-pair layout
  for (int idx = threadIdx.x; idx < (K << 4); idx += blockDim.x) {
    int p = idx >> 5;                        // K-pair index
    int n = (idx >> 1) & 15;                 // column within tile
    int j = idx & 1;                         // element within pair
    Bs[idx] = B[(size_t)((p << 1) + j) * Nout + col0 + n];
  }
  __syncthreads();

  const int rowTile = rowGroup * WAVES_PER_BLOCK + wave;
  if ((rowTile << 4) >= M) return;           // wave-uniform (after barrier)
  const int row0 = rowTile << 4;

  const int m    = lane & 15;                // A row / C column within tile
  const int half = lane >> 4;                // K sub-pair selector

  const float*  aRow = A + (size_t)(row0 + m) * K + (half << 1);
  const float* bBase = Bs + (half << 5) + (m << 1);

  v8f c = {};
#pragma unroll 4
  for (int k = 0; k < K; k += 4) {
    v2f a = *(const v2f*)(aRow + k);         // contiguous K pair from global
    v2f b = *(const v2f*)(bBase + (k << 3)); // one ds_load_b64 (offset k*8 floats)
    c = __builtin_amdgcn_wmma_f32_16x16x4_f32(
        /*neg_a=*/false, a, /*neg_b=*/false, b,
        /*c_mod=*/(short)0, c, /*reuse_a=*/false, /*reuse_b=*/false);
  }

  float* out = C + (size_t)(row0 + (half << 3)) * Nout + col0 + m;
#pragma unroll
  for (int v = 0; v < 8; ++v) out[(size_t)v * Nout] = c[v];
}

// ---------------------------------------------------------------------------
// 3) layer-1 edge scatter: 32 threads/edge, float4 chunk each (H1=128)
// ---------------------------------------------------------------------------
__global__ void k_scatter1(const float* __restrict__ h,
                           const int* __restrict__ src,
                           const int* __restrict__ dst,
                           const float* __restrict__ dinv,
                           float* __restrict__ agg, long long nE) {
  long long idx = (long long)blockIdx.x * blockDim.x + threadIdx.x;
  long long total = nE << 5;                 // E * 32 chunks
  if (idx >= total) return;
  long long e = idx >> 5;
  int c4 = (int)(idx & 31) << 2;             // channel offset 0..124
  // stream-ahead prefetch of the index arrays (global_prefetch_b8)
  if (e + 2048 < nE) {
    __builtin_prefetch(src + e + 2048, 0, 0);
    __builtin_prefetch(dst + e + 2048, 0, 0);
  }
  int s = src[e], d = dst[e];
  float coef = dinv[s] * dinv[d];
  const float4 hv = *(const float4*)(h + (size_t)s * 128 + c4);
  float* ap = agg + (size_t)d * 128 + c4;
  atomicAdd(ap + 0, hv.x * coef);
  atomicAdd(ap + 1, hv.y * coef);
  atomicAdd(ap + 2, hv.z * coef);
  atomicAdd(ap + 3, hv.w * coef);
}

// ---------------------------------------------------------------------------
// 4) finalize layer 1: hidden = relu(agg + h*dinv^2 + b1), in place in agg1
// ---------------------------------------------------------------------------
__global__ void k_fin1(float* __restrict__ agg, const float* __restrict__ h,
                       const float* __restrict__ dinv,
                       const float* __restrict__ b1, long long total) {
  long long i = (long long)blockIdx.x * blockDim.x + threadIdx.x;
  if (i >= total) return;
  int node = (int)(i >> 7);                  // / 128
  int c    = (int)(i & 127);
  float di = dinv[node];
  float v  = agg[i] + h[i] * (di * di) + b1[c];
  agg[i] = v > 0.0f ? v : 0.0f;
}

// ---------------------------------------------------------------------------
// 5) fused layer-2/3 edge scatter: lanes 0-15 -> mu chunks, 16-31 -> logvar
//    (H2=64 -> 16 float4 chunks per matrix, 32 per edge). Atomics into d_out.
// ---------------------------------------------------------------------------
__global__ void k_scatter23(const float* __restrict__ h2,
                            const float* __restrict__ h3,
                            const int* __restrict__ src,
                            const int* __restrict__ dst,
                            const float* __restrict__ dinv,
                            float* __restrict__ out, int n_nodes,
                            long long nE) {
  long long idx = (long long)blockIdx.x * blockDim.x + threadIdx.x;
  long long total = nE << 5;
  if (idx >= total) return;
  long long e = idx >> 5;
  int t = (int)(idx & 31);
  int which = t >> 4;                        // 0 = mu, 1 = logvar
  int c4 = (t & 15) << 2;                    // channel offset 0..60
  if (e + 2048 < nE) {
    __builtin_prefetch(src + e + 2048, 0, 0);
    __builtin_prefetch(dst + e + 2048, 0, 0);
  }
  int s = src[e], d = dst[e];
  float coef = dinv[s] * dinv[d];
  const float* h = which ? h3 : h2;
  float* agg = out + (size_t)which * (size_t)n_nodes * 64;
  const float4 hv = *(const float4*)(h + (size_t)s * 64 + c4);
  float* ap = agg + (size_t)d * 64 + c4;
  atomicAdd(ap + 0, hv.x * coef);
  atomicAdd(ap + 1, hv.y * coef);
  atomicAdd(ap + 2, hv.z * coef);
  atomicAdd(ap + 3, hv.w * coef);
}

// ---------------------------------------------------------------------------
// 6) finalize mu & logvar: out += h*dinv^2 + b   (no activation)
// ---------------------------------------------------------------------------
__global__ void k_fin23(float* __restrict__ out,
                        const float* __restrict__ h2,
                        const float* __restrict__ h3,
                        const float* __restrict__ dinv,
                        const float* __restrict__ b2,
                        const float* __restrict__ b3,
                        int n_nodes, long long total) {
  long long i = (long long)blockIdx.x * blockDim.x + threadIdx.x;
  if (i >= total) return;                    // total = 2*N*64
  long long half = (long long)n_nodes * 64;
  int which = i >= half;
  long long local = i - (which ? half : 0);
  int node = (int)(local >> 6);
  int c    = (int)(local & 63);
  float di = dinv[node];
  const float* h = which ? h3 : h2;
  const float* b = which ? b3 : b2;
  out[i] += h[local] * (di * di) + b[c];
}

// ---------------------------------------------------------------------------
// launch
// ---------------------------------------------------------------------------
extern "C" void kernel_launch(void* const* d_in, const int* in_sizes, int n_in,
                              void* d_out, int out_size, void* d_ws, size_t ws_size,
                              hipStream_t stream) {
  const float* x  = (const float*)d_in[0];
  const float* w1 = (const float*)d_in[1];
  const float* b1 = (const float*)d_in[2];
  const float* w2 = (const float*)d_in[3];
  const float* b2 = (const float*)d_in[4];
  const float* w3 = (const float*)d_in[5];
  const float* b3 = (const float*)d_in[6];
  const int*   ei = (const int*)d_in[7];

  const int F_IN = 256, H1 = 128, H2 = 64;
  const int Nn = in_sizes[0] / F_IN;                // 100000
  const long long Ee = (long long)in_sizes[7] / 2;  // 1600000
  const int* src = ei;
  const int* dst = ei + Ee;

  float* out = (float*)d_out;                   // [N*64 mu | N*64 logvar]

  // workspace layout (floats)
  float* dinv = (float*)d_ws;                   // N       (deg -> dinv_sqrt)
  float* h1t  = dinv + Nn;                      // N*128   x @ w1
  float* agg1 = h1t + (size_t)Nn * H1;          // N*128   agg -> hidden
  float* h2t  = agg1 + (size_t)Nn * H1;         // N*64    hidden @ w2
  float* h3t  = h2t + (size_t)Nn * H2;          // N*64    hidden @ w3

  const int B = 256;
  long long nH1 = (long long)Nn * H1;           // 12.8M (also == out_size)
  long long nEdgeWork = Ee << 5;                // 51.2M

  const int rowTiles  = Nn / 16;                                   // 6250
  const int rowGroups = (rowTiles + WAVES_PER_BLOCK - 1) / WAVES_PER_BLOCK; // 625

  // 0) init
  k_init<<<(int)((nH1 + B - 1) / B), B, 0, stream>>>(agg1, out, dinv, Nn, nH1);
  // 1) degree + rsqrt
  k_deg<<<(int)((Ee + B - 1) / B), B, 0, stream>>>(dst, dinv, Ee);
  k_rsqrt<<<(Nn + B - 1) / B, B, 0, stream>>>(dinv, Nn);
  // 2) h1t = x @ w1   (625 row groups x 8 col tiles)
  k_gemm_wmma<<<rowGroups * (H1 / 16), GEMM_BLOCK, 0, stream>>>(x, w1, h1t, Nn, F_IN, H1);
  // 3) edge scatter layer 1
  k_scatter1<<<(int)((nEdgeWork + B - 1) / B), B, 0, stream>>>(h1t, src, dst, dinv, agg1, Ee);
  // 4) hidden = relu(agg1 + h1t*dinv^2 + b1)
  k_fin1<<<(int)((nH1 + B - 1) / B), B, 0, stream>>>(agg1, h1t, dinv, b1, nH1);
  // 5) h2t/h3t = hidden @ w2 / w3  (625 x 4 col tiles each)
  k_gemm_wmma<<<rowGroups * (H2 / 16), GEMM_BLOCK, 0, stream>>>(agg1, w2, h2t, Nn, H1, H2);
  k_gemm_wmma<<<rowGroups * (H2 / 16), GEMM_BLOCK, 0, stream>>>(agg1, w3, h3t, Nn, H1, H2);
  // 6) fused edge scatter layers 2+3 (atomics straight into d_out)
  k_scatter23<<<(int)((nEdgeWork + B - 1) / B), B, 0, stream>>>(h2t, h3t, src, dst, dinv, out, Nn, Ee);
  // 7) finalize mu / logvar
  k_fin23<<<(int)((nH1 + B - 1) / B), B, 0, stream>>>(out, h2t, h3t, dinv, b2, b3, Nn, nH1);
}